// DeformConv2d_12008728560143
// MI455X (gfx1250) — compile-verified
//
#include <hip/hip_runtime.h>
#include <hip/hip_bf16.h>
#include <math.h>

typedef __attribute__((ext_vector_type(2))) float v2f;
typedef __attribute__((ext_vector_type(8))) float v8f;

static constexpr int B_    = 2;
static constexpr int C_    = 64;
static constexpr int H_    = 128;
static constexpr int W_    = 128;
static constexpr int OC_   = 64;
static constexpr int NS_   = 9;          // 3x3 samples
static constexpr int OFFC_ = 18;         // 2*NS
static constexpr int KT_   = C_ * NS_;   // 576 = GEMM K

// ---------------------------------------------------------------------------
// Kernel A: offset = conv3x3(x, w_p) + b_p   (B,18,H,W), pad=1
// ---------------------------------------------------------------------------
__global__ void __launch_bounds__(256) k_offset_conv(
    const float* __restrict__ x, const float* __restrict__ wp,
    const float* __restrict__ bp, float* __restrict__ offs) {
  int idx = blockIdx.x * blockDim.x + threadIdx.x;
  if (idx >= B_ * OFFC_ * H_ * W_) return;
  int w  = idx % W_;
  int h  = (idx / W_) % H_;
  int co = (idx / (W_ * H_)) % OFFC_;
  int b  = idx / (W_ * H_ * OFFC_);
  float acc = bp[co];
  const float* xb = x  + (size_t)b * C_ * H_ * W_;
  const float* wc = wp + (size_t)co * C_ * 9;
  for (int c = 0; c < C_; ++c) {
    const float* pl = xb + c * H_ * W_;
    const float* wk = wc + c * 9;
#pragma unroll
    for (int ki = 0; ki < 3; ++ki) {
      int ih = h + ki - 1;
      if (ih < 0 || ih >= H_) continue;
#pragma unroll
      for (int kj = 0; kj < 3; ++kj) {
        int iw = w + kj - 1;
        if (iw < 0 || iw >= W_) continue;
        acc = fmaf(wk[ki * 3 + kj], pl[ih * W_ + iw], acc);
      }
    }
  }
  offs[idx] = acc;
}

// fetch from a virtually zero-padded (pad=1) H_ x W_ plane at padded coords
__device__ __forceinline__ float fetch_pad(const float* plane, int qx, int qy) {
  if (qx >= 1 && qx <= H_ && qy >= 1 && qy <= W_)
    return plane[(qx - 1) * W_ + (qy - 1)];
  return 0.f;
}

// Reference-exact bilinear: clip indices/coords to [0, clipmax], weights may
// go negative out of range (matches the jnp formulation).
__device__ __forceinline__ float bilin(const float* plane, float px, float py,
                                       float clipmax) {
  float fx = floorf(px), fy = floorf(py);
  int x0 = (int)fminf(fmaxf(fx,       0.f), clipmax);
  int y0 = (int)fminf(fmaxf(fy,       0.f), clipmax);
  int x1 = (int)fminf(fmaxf(fx + 1.f, 0.f), clipmax);
  int y1 = (int)fminf(fmaxf(fy + 1.f, 0.f), clipmax);
  float pxc = fminf(fmaxf(px, 0.f), clipmax);
  float pyc = fminf(fmaxf(py, 0.f), clipmax);
  float glt = (1.f + ((float)x0 - pxc)) * (1.f + ((float)y0 - pyc));
  float grb = (1.f - ((float)x1 - pxc)) * (1.f - ((float)y1 - pyc));
  float glb = (1.f + ((float)x0 - pxc)) * (1.f - ((float)y1 - pyc));
  float grt = (1.f - ((float)x1 - pxc)) * (1.f + ((float)y0 - pyc));
  return glt * fetch_pad(plane, x0, y0) + grb * fetch_pad(plane, x1, y1) +
         glb * fetch_pad(plane, x0, y1) + grt * fetch_pad(plane, x1, y0);
}

// ---------------------------------------------------------------------------
// Kernel B: depth bilinear -> depth_weight / m, modulated positions Px,Py, Mw
// ---------------------------------------------------------------------------
__global__ void __launch_bounds__(256) k_sample_prep(
    const float* __restrict__ depth, const float* __restrict__ offs,
    float* __restrict__ Px, float* __restrict__ Py, float* __restrict__ Mw) {
  int idx = blockIdx.x * blockDim.x + threadIdx.x;
  if (idx >= B_ * H_ * W_) return;
  int w = idx % W_;
  int h = (idx / W_) % H_;
  int b = idx / (H_ * W_);
  float d0 = depth[idx];                       // depth is (B,1,H,W)
  const float* db = depth + (size_t)b * H_ * W_;
#pragma unroll
  for (int k = 0; k < NS_; ++k) {
    float offx = offs[(((size_t)b * OFFC_ + k)       * H_ + h) * W_ + w];
    float offy = offs[(((size_t)b * OFFC_ + NS_ + k) * H_ + h) * W_ + w];
    float pnx = (float)(k / 3 - 1);
    float pny = (float)(k % 3 - 1);
    float px = offx + pnx + (float)(h + 1);
    float py = offy + pny + (float)(w + 1);
    // depth sampled with clip bound H-1 (=127) per reference
    float dq = bilin(db, px, py, (float)(H_ - 1));
    float dd = fabsf(d0 - dq);
    float dwk = expf(-4.f * dd) + 0.25f;       // depth_weight
    float mk  = expf(-dd);                     // modulation
    Px[(size_t)idx * NS_ + k] = offx * dwk + pnx + (float)(h + 1);
    Py[(size_t)idx * NS_ + k] = offy * dwk + pny + (float)(w + 1);
    Mw[(size_t)idx * NS_ + k] = mk;
  }
}

// ---------------------------------------------------------------------------
// Kernel C: fused gather + GEMM via V_WMMA_F32_16X16X4_F32.
// Block = 128 threads (4 waves) owns 16 pixels; LDS holds Xcol tile 576x16.
// Wave w computes output channels [16w, 16w+16) for those 16 pixels.
// ---------------------------------------------------------------------------
__global__ void __launch_bounds__(128) k_deform_gemm(
    const float* __restrict__ x, const float* __restrict__ wconv,
    const float* __restrict__ Px, const float* __restrict__ Py,
    const float* __restrict__ Mw, float* __restrict__ out) {
  __shared__ float Bt[KT_ * 16];               // 36 KB
  const int tid  = threadIdx.x;
  const int pix0 = blockIdx.x * 16;

  // ---- phase 1: build Xcol tile [K=c*9+k][n] in LDS ----
  for (int p = tid; p < 16 * NS_; p += 128) {
    int n = p / NS_;
    int k = p % NS_;
    int g = pix0 + n;
    int b = g / (H_ * W_);
    float px = Px[(size_t)g * NS_ + k];
    float py = Py[(size_t)g * NS_ + k];
    float m  = Mw[(size_t)g * NS_ + k];
    float fx = floorf(px), fy = floorf(py);
    const float clipmax = (float)(H_ + 1);     // Hp-1 = 129 for x sampling
    int x0 = (int)fminf(fmaxf(fx,       0.f), clipmax);
    int y0 = (int)fminf(fmaxf(fy,       0.f), clipmax);
    int x1 = (int)fminf(fmaxf(fx + 1.f, 0.f), clipmax);
    int y1 = (int)fminf(fmaxf(fy + 1.f, 0.f), clipmax);
    float pxc = fminf(fmaxf(px, 0.f), clipmax);
    float pyc = fminf(fmaxf(py, 0.f), clipmax);
    float glt = m * (1.f + ((float)x0 - pxc)) * (1.f + ((float)y0 - pyc));
    float grb = m * (1.f - ((float)x1 - pxc)) * (1.f - ((float)y1 - pyc));
    float glb = m * (1.f + ((float)x0 - pxc)) * (1.f - ((float)y1 - pyc));
    float grt = m * (1.f - ((float)x1 - pxc)) * (1.f + ((float)y0 - pyc));
    bool ilt = (x0 >= 1) && (x0 <= H_) && (y0 >= 1) && (y0 <= W_);
    bool irb = (x1 >= 1) && (x1 <= H_) && (y1 >= 1) && (y1 <= W_);
    bool ilb = (x0 >= 1) && (x0 <= H_) && (y1 >= 1) && (y1 <= W_);
    bool irt = (x1 >= 1) && (x1 <= H_) && (y0 >= 1) && (y0 <= W_);
    int olt = (x0 - 1) * W_ + (y0 - 1);
    int orb = (x1 - 1) * W_ + (y1 - 1);
    int olb = (x0 - 1) * W_ + (y1 - 1);
    int ort = (x1 - 1) * W_ + (y0 - 1);
    const float* xb = x + (size_t)b * C_ * H_ * W_;
    for (int c = 0; c < C_; ++c) {
      const float* pl = xb + c * H_ * W_;
      float v = 0.f;
      if (ilt) v = fmaf(glt, pl[olt], v);
      if (irb) v = fmaf(grb, pl[orb], v);
      if (ilb) v = fmaf(glb, pl[olb], v);
      if (irt) v = fmaf(grt, pl[ort], v);
      Bt[(c * NS_ + k) * 16 + n] = v;          // Xcol[K][n]
    }
  }
  __syncthreads();

  // ---- phase 2: 144 x v_wmma_f32_16x16x4_f32 per wave ----
  const int lane = tid & 31;
  const int wave = tid >> 5;
  const int half = lane >> 4;                  // 0: K+0/1, 1: K+2/3
  const int l16  = lane & 15;
  const int ocb  = wave * 16;
  v8f acc = {};
  for (int kk = 0; kk < KT_; kk += 4) {
    int ka = kk + half * 2;
    v2f a, bb;
    const float* wr = wconv + (size_t)(ocb + l16) * KT_ + ka;  // A: row=oc
    a.x  = wr[0];
    a.y  = wr[1];
    bb.x = Bt[ka * 16 + l16];                                  // B: col=pixel
    bb.y = Bt[(ka + 1) * 16 + l16];
    acc = __builtin_amdgcn_wmma_f32_16x16x4_f32(
        /*neg_a=*/false, a, /*neg_b=*/false, bb,
        /*c_mod=*/(short)0, acc, /*reuse_a=*/false, /*reuse_b=*/false);
  }

  // ---- writeback per 16x16 f32 C/D layout ----
  int g = pix0 + l16;
  int w = g % W_;
  int h = (g / W_) % H_;
  int b = g / (H_ * W_);
#pragma unroll
  for (int r = 0; r < 8; ++r) {
    int oc = ocb + half * 8 + r;
    out[(((size_t)b * OC_ + oc) * H_ + h) * W_ + w] = acc[r];
  }
}

// ---------------------------------------------------------------------------
extern "C" void kernel_launch(void* const* d_in, const int* in_sizes, int n_in,
                              void* d_out, int out_size, void* d_ws, size_t ws_size,
                              hipStream_t stream) {
  const float* x     = (const float*)d_in[0];
  const float* depth = (const float*)d_in[1];
  const float* w_p   = (const float*)d_in[2];
  const float* b_p   = (const float*)d_in[3];
  const float* wconv = (const float*)d_in[4];
  float* out = (float*)d_out;

  float* ws   = (float*)d_ws;
  float* offs = ws;                                  // B*18*H*W   = 589824 f
  float* Px   = offs + (size_t)B_ * OFFC_ * H_ * W_; // B*H*W*9    = 294912 f
  float* Py   = Px   + (size_t)B_ * H_ * W_ * NS_;
  float* Mw   = Py   + (size_t)B_ * H_ * W_ * NS_;   // total ~5.9 MB

  int totA = B_ * OFFC_ * H_ * W_;
  k_offset_conv<<<(totA + 255) / 256, 256, 0, stream>>>(x, w_p, b_p, offs);

  int totB = B_ * H_ * W_;
  k_sample_prep<<<(totB + 255) / 256, 256, 0, stream>>>(depth, offs, Px, Py, Mw);

  int nblk = (B_ * H_ * W_) / 16;                    // 2048 blocks, 128 thr
  k_deform_gemm<<<nblk, 128, 0, stream>>>(x, wconv, Px, Py, Mw, out);
}